// SbS_29764123361671
// MI455X (gfx1250) — compile-verified
//
#include <hip/hip_runtime.h>
#include <stdint.h>

#ifndef __has_builtin
#define __has_builtin(x) 0
#endif

#define B_   16
#define T_   128
#define X_   16
#define Y_   16
#define IN_  256
#define H_   64

// LDS layout (floats):
//   [0,16384)      weights [IN][H]              64 KB
//   [16384,16512)  eps_t   [T]                  0.5 KB
//   [16512,18560)  spikes  [T][Y] (int)         8 KB   (reused for output transpose)
//   [18560,22656)  eps_xy  slice [IN][Y] at x   16 KB
#define SMEM_FLOATS 22656
#define SMEM_BYTES  (SMEM_FLOATS * 4)

// ---------------- CDNA5 async global->LDS staging (guarded) ----------------
#if defined(__HIP_DEVICE_COMPILE__) && \
    __has_builtin(__builtin_amdgcn_global_load_async_to_lds_b128)
#define SBS_ASYNC 1
#else
#define SBS_ASYNC 0
#endif

#if SBS_ASYNC
typedef int v4i __attribute__((ext_vector_type(4)));
typedef __attribute__((address_space(1))) v4i* g_v4i_p;  // global int4*
typedef __attribute__((address_space(3))) v4i* l_v4i_p;  // LDS int4* (32-bit)

__device__ __forceinline__ void cp_b128(void* lds_dst, const void* g_src) {
  // Copies 16 bytes from global g_src to LDS lds_dst, tracked by ASYNCcnt.
  // Generic->AS casts via integer; flat LDS addr low 32 bits == LDS byte offset.
  __builtin_amdgcn_global_load_async_to_lds_b128(
      (g_v4i_p)(uintptr_t)g_src,
      (l_v4i_p)(unsigned int)(uintptr_t)lds_dst,
      /*imm offset*/ 0, /*cpol*/ 0);
}
__device__ __forceinline__ void async_wait_all() {
#if __has_builtin(__builtin_amdgcn_s_wait_asynccnt)
  __builtin_amdgcn_s_wait_asynccnt(0);
#else
  asm volatile("s_wait_asynccnt 0" ::: "memory");
#endif
}
#else
__device__ __forceinline__ void cp_b128(void* lds_dst, const void* g_src) {
  *(float4*)lds_dst = *(const float4*)g_src;
}
__device__ __forceinline__ void async_wait_all() {}
#endif

__device__ __forceinline__ float fast_rcp(float x) {
#if defined(__HIP_DEVICE_COMPILE__) && __has_builtin(__builtin_amdgcn_rcpf)
  return __builtin_amdgcn_rcpf(x);
#else
  return 1.0f / x;
#endif
}

// ---------------------------------------------------------------------------
// One block = one (b,x) slice: 16 pixels (y = 0..15), 256 threads (8 waves).
// Lane j (= tid & 15) of pixel y (= tid >> 4) holds h[4j..4j+3].
// Two pixels per wave32; the H=64 dot reduces over the 16-lane half-wave
// with a shfl_xor tree (masks 1,2,4,8 stay inside the half-wave).
// ---------------------------------------------------------------------------
__global__ void __launch_bounds__(256)
sbs_hdyn_kernel(const int*   __restrict__ spikes,   // [B,T,X,Y]
                const float* __restrict__ eps_xy,   // [IN,X,Y]
                const float* __restrict__ eps_t,    // [T]
                const float* __restrict__ weights,  // [IN,H]
                const float* __restrict__ h_init,   // [H]
                const float* __restrict__ fo_p,     // [1]
                float*       __restrict__ out) {    // [B,H,X,Y]
  extern __shared__ float smem[];
  float* w_lds   = smem;                   // 16384 f
  float* et_lds  = smem + 16384;           // 128 f
  int*   spk_lds = (int*)(smem + 16512);   // 2048 i
  float* eps_lds = smem + 18560;           // 4096 f

  const int tid = threadIdx.x;
  const int b   = blockIdx.x >> 4;
  const int x   = blockIdx.x & 15;
  const int y   = tid >> 4;   // pixel within block
  const int j   = tid & 15;   // h-chunk lane within pixel

  // ---- stage everything the T-loop touches into LDS (ASYNCcnt-tracked) ----
  #pragma unroll
  for (int it = 0; it < 16; ++it) {        // weights: 4096 x b128
    const int c = tid + 256 * it;
    cp_b128(&w_lds[c * 4], &weights[c * 4]);
  }
  if (tid < 32)                            // eps_t: 32 x b128
    cp_b128(&et_lds[tid * 4], &eps_t[tid * 4]);
  #pragma unroll
  for (int it = 0; it < 2; ++it) {         // spike sequence: 512 x b128
    const int c = tid + 256 * it;
    const int t = c >> 2, y0 = (c & 3) * 4;
    cp_b128(&spk_lds[t * 16 + y0],
            &spikes[((b * T_ + t) * X_ + x) * Y_ + y0]);
  }
  #pragma unroll
  for (int it = 0; it < 4; ++it) {         // eps_xy[:,x,:] slice: 1024 x b128
    const int c = tid + 256 * it;
    const int i = c >> 2, y0 = (c & 3) * 4;
    cp_b128(&eps_lds[i * 16 + y0],
            &eps_xy[(i * X_ + x) * Y_ + y0]);
  }
  async_wait_all();     // this wave's async LDS writes landed
  __syncthreads();      // all waves' writes visible

  const float fov  = fo_p[0];
  const float fo_h = fov * (1.0f / (float)H_);
  const float c1fo = 1.0f + fov;

  float4 h = *(const float4*)&h_init[j * 4];

  #pragma unroll 4
  for (int t = 0; t < T_; ++t) {
    const int    sp = spk_lds[t * 16 + y];                     // LDS broadcast
    const float4 w  = *(const float4*)&w_lds[sp * H_ + j * 4]; // ds_load_b128
    float part = h.x * w.x + h.y * w.y + h.z * w.z + h.w * w.w;
    part += __shfl_xor(part, 1);
    part += __shfl_xor(part, 2);
    part += __shfl_xor(part, 4);
    part += __shfl_xor(part, 8);
    const float s   = part;                                    // h . w
    const float eps = eps_lds[sp * 16 + y] * et_lds[t];
    if (s > 1e-10f) {                                          // uniform per pixel
      const float scale = eps * fast_rcp(s);
      const float rden  = fast_rcp(1.0f + eps * c1fo);
      const float add   = eps * fo_h;
      // h <- (h*(1 + scale*w) + eps*fo/H) / (1 + eps*(1+fo))
      h.x = (h.x * (1.0f + scale * w.x) + add) * rden;
      h.y = (h.y * (1.0f + scale * w.y) + add) * rden;
      h.z = (h.z * (1.0f + scale * w.z) + add) * rden;
      h.w = (h.w * (1.0f + scale * w.w) + add) * rden;
    }
  }

  // ---- transpose through LDS for coalesced [B,H,X,Y] stores ----
  __syncthreads();                       // done reading spk_lds; reuse it
  float* out_lds = (float*)spk_lds;      // 1024 floats = this block's tile
  out_lds[(j * 4 + 0) * 16 + y] = h.x;
  out_lds[(j * 4 + 1) * 16 + y] = h.y;
  out_lds[(j * 4 + 2) * 16 + y] = h.z;
  out_lds[(j * 4 + 3) * 16 + y] = h.w;
  __syncthreads();

  #pragma unroll
  for (int it = 0; it < 4; ++it) {
    const int idx = tid + it * 256;
    const int hh = idx >> 4, yy = idx & 15;
    out[((b * H_ + hh) * X_ + x) * Y_ + yy] = out_lds[idx];
  }
}

extern "C" void kernel_launch(void* const* d_in, const int* in_sizes, int n_in,
                              void* d_out, int out_size, void* d_ws, size_t ws_size,
                              hipStream_t stream) {
  (void)in_sizes; (void)n_in; (void)d_ws; (void)ws_size; (void)out_size;
  // setup_inputs order: 0 input(unused fwd), 1 spikes, 2 epsilon_xy,
  //                     3 epsilon_t_0, 4 weights, 5 h_initial, 6 forgetting_offset
  const int*   spikes  = (const int*)d_in[1];
  const float* eps_xy  = (const float*)d_in[2];
  const float* eps_tt  = (const float*)d_in[3];
  const float* weights = (const float*)d_in[4];
  const float* h_init  = (const float*)d_in[5];
  const float* fo      = (const float*)d_in[6];
  float* out = (float*)d_out;

  // ~90 KB dynamic LDS (gfx1250 WGP has 320 KB; 3 workgroups/WGP).
  (void)hipFuncSetAttribute((const void*)sbs_hdyn_kernel,
                            hipFuncAttributeMaxDynamicSharedMemorySize,
                            SMEM_BYTES);
  sbs_hdyn_kernel<<<dim3(B_ * X_), dim3(256), SMEM_BYTES, stream>>>(
      spikes, eps_xy, eps_tt, weights, h_init, fo, out);
}